// MomentumModel_76209899700576
// MI455X (gfx1250) — compile-verified
//
#include <hip/hip_runtime.h>
#include <hip/hip_bf16.h>
#include <math.h>

#define T_ 8
#define N_ 50000
#define F_ 64
#define H_ 32
#define L_ 64
#define E_ 800000

typedef __attribute__((ext_vector_type(2))) float v2f;
typedef __attribute__((ext_vector_type(8))) float v8f;

// ---------------------------------------------------------------------------
// 0. init: deg = 1.0 (self loop), agg = 0, seq = 0
// ---------------------------------------------------------------------------
__global__ __launch_bounds__(256) void init_kernel(float* __restrict__ deg,
                                                   float* __restrict__ agg,
                                                   float* __restrict__ seq) {
    long idx = (long)blockIdx.x * blockDim.x + threadIdx.x;
    long tot = (long)T_ * N_ * H_;
    if (idx < tot)            agg[idx] = 0.0f;
    if (idx < (long)T_ * N_)  deg[idx] = 1.0f;       // self-loop contribution
    if (idx < T_ * H_)        seq[idx] = 0.0f;
}

// ---------------------------------------------------------------------------
// 1. degree accumulation: deg[t, dst] += 1 per edge
// ---------------------------------------------------------------------------
__global__ __launch_bounds__(256) void degree_kernel(const int* __restrict__ ei,
                                                     float* __restrict__ deg) {
    long idx = (long)blockIdx.x * blockDim.x + threadIdx.x;
    if (idx >= (long)T_ * E_) return;
    int t = (int)(idx / E_);
    int e = (int)(idx % E_);
    int dst = ei[(long)t * 2 * E_ + E_ + e];
    atomicAdd(&deg[(long)t * N_ + dst], 1.0f);
}

// ---------------------------------------------------------------------------
// 2. deg -> rsqrt(deg) in place (deg >= 1 always)
// ---------------------------------------------------------------------------
__global__ __launch_bounds__(256) void rsqrt_kernel(float* __restrict__ deg) {
    long idx = (long)blockIdx.x * blockDim.x + threadIdx.x;
    if (idx < (long)T_ * N_) deg[idx] = rsqrtf(deg[idx]);
}

// ---------------------------------------------------------------------------
// 3. XW = X @ W  via V_WMMA_F32_16X16X4_F32 (fp32 matrix pipe, wave32)
//    X: [M=T*N, 64] row-major, W: [64, 32] row-major, XW: [M, 32]
//    One wave computes a 16x32 strip: 2 C-tiles, 16 K-steps of 4 -> 32 WMMAs.
// ---------------------------------------------------------------------------
__global__ __launch_bounds__(256) void gemm_xw_wmma(const float* __restrict__ X,
                                                    const float* __restrict__ W,
                                                    float* __restrict__ XW,
                                                    int Mtotal) {
    int wave = (int)(((long)blockIdx.x * blockDim.x + threadIdx.x) >> 5);
    int lane = threadIdx.x & 31;
    int m0 = wave * 16;
    if (m0 >= Mtotal) return;                 // wave-uniform: EXEC stays all-1s

    int mrow = lane & 15;                     // M index for A (both lane halves)
    int koff = (lane >> 4) * 2;               // lanes 16-31 hold K+2, K+3
    int ncol = lane & 15;                     // N index for B (both lane halves)

    const float* xr = X + (long)(m0 + mrow) * F_;

    v8f c0 = {}; v8f c1 = {};
#pragma unroll
    for (int k = 0; k < F_; k += 4) {
        int kb = k + koff;
        v2f a;  a.x  = xr[kb];               a.y  = xr[kb + 1];
        v2f b0; b0.x = W[(kb)     * H_ + ncol];      b0.y = W[(kb + 1) * H_ + ncol];
        v2f b1; b1.x = W[(kb) * H_ + 16 + ncol];     b1.y = W[(kb + 1) * H_ + 16 + ncol];
        c0 = __builtin_amdgcn_wmma_f32_16x16x4_f32(false, a, false, b0, (short)0, c0, false, false);
        c1 = __builtin_amdgcn_wmma_f32_16x16x4_f32(false, a, false, b1, (short)0, c1, false, false);
    }

    // C/D layout: VGPR r -> M=r (lanes 0-15), M=r+8 (lanes 16-31); N = lane%16
    int mhi = (lane >> 4) * 8;
#pragma unroll
    for (int r = 0; r < 8; r++) {
        long orow = (long)(m0 + mhi + r) * H_;
        XW[orow + ncol]      = c0[r];
        XW[orow + 16 + ncol] = c1[r];
    }
}

// ---------------------------------------------------------------------------
// 4. edge scatter: one wave per edge, lane = feature channel (H_ == wave32)
//    agg[t,dst,h] += dinv[t,src]*dinv[t,dst] * xw[t,src,h]
// ---------------------------------------------------------------------------
__global__ __launch_bounds__(256) void scatter_kernel(const int* __restrict__ ei,
                                                      const float* __restrict__ dinv,
                                                      const float* __restrict__ xw,
                                                      float* __restrict__ agg) {
    long tid = (long)blockIdx.x * blockDim.x + threadIdx.x;
    int  h   = (int)(tid & 31);
    long eid = tid >> 5;
    if (eid >= (long)T_ * E_) return;
    int t = (int)(eid / E_);
    int e = (int)(eid % E_);
    int src = ei[(long)t * 2 * E_ + e];        // wave-uniform load
    int dst = ei[(long)t * 2 * E_ + E_ + e];   // wave-uniform load
    long si = (long)t * N_ + src;
    long di = (long)t * N_ + dst;
    float nrm = dinv[si] * dinv[di];
    atomicAdd(&agg[di * H_ + h], nrm * xw[si * H_ + h]);
}

// ---------------------------------------------------------------------------
// 5. relu + mean-pool over nodes (adds self-loop term + bias here)
//    grid: (blocks, T); block = 8 node-rows x 32 channels; LDS reduce
// ---------------------------------------------------------------------------
__global__ __launch_bounds__(256) void pool_kernel(const float* __restrict__ agg,
                                                   const float* __restrict__ xw,
                                                   const float* __restrict__ dinv,
                                                   const float* __restrict__ conv_b,
                                                   float* __restrict__ seq) {
    int t = blockIdx.y;
    int h = threadIdx.x & 31;
    int r = threadIdx.x >> 5;                 // 0..7
    float bh = conv_b[h];
    float acc = 0.0f;
    for (int nd = blockIdx.x * 8 + r; nd < N_; nd += gridDim.x * 8) {
        long i = (long)t * N_ + nd;
        float di = dinv[i];
        float v = agg[i * H_ + h] + di * di * xw[i * H_ + h] + bh;
        acc += fmaxf(v, 0.0f);
    }
    __shared__ float red[8][H_];
    red[r][h] = acc;
    __syncthreads();
    if (r == 0) {
        float s = 0.0f;
#pragma unroll
        for (int j = 0; j < 8; j++) s += red[j][h];
        atomicAdd(&seq[t * H_ + h], s * (1.0f / (float)N_));
    }
}

// ---------------------------------------------------------------------------
// 6. LSTM (8 steps, batch 1) + FC head. Single block, thread = gate row.
//    torch gate order i,f,g,o.
// ---------------------------------------------------------------------------
__global__ __launch_bounds__(256) void lstm_kernel(const float* __restrict__ seq,
                                                   const float* __restrict__ w_ih,
                                                   const float* __restrict__ w_hh,
                                                   const float* __restrict__ b_ih,
                                                   const float* __restrict__ b_hh,
                                                   const float* __restrict__ fc_w,
                                                   const float* __restrict__ fc_b,
                                                   float* __restrict__ out) {
    __shared__ float hs[L_], cs[L_], gates[4 * L_], sq[T_ * H_];
    int tid = threadIdx.x;
    if (tid < L_) { hs[tid] = 0.0f; cs[tid] = 0.0f; }
    sq[tid] = seq[tid];                        // 256 == T_*H_
    __syncthreads();

    for (int t = 0; t < T_; t++) {
        float g = b_ih[tid] + b_hh[tid];
        const float* wi = w_ih + tid * H_;
        const float* xs = sq + t * H_;
#pragma unroll
        for (int k = 0; k < H_; k++) g += wi[k] * xs[k];
        const float* wh = w_hh + tid * L_;
#pragma unroll
        for (int k = 0; k < L_; k++) g += wh[k] * hs[k];
        gates[tid] = g;
        __syncthreads();
        if (tid < L_) {
            float ig = 1.0f / (1.0f + __expf(-gates[tid]));
            float fg = 1.0f / (1.0f + __expf(-gates[L_ + tid]));
            float gg = tanhf(gates[2 * L_ + tid]);
            float og = 1.0f / (1.0f + __expf(-gates[3 * L_ + tid]));
            float cn = fg * cs[tid] + ig * gg;
            cs[tid] = cn;
            hs[tid] = og * tanhf(cn);
        }
        __syncthreads();
    }
    if (tid == 0) {
        float s = fc_b[0];
        for (int j = 0; j < L_; j++) s += hs[j] * fc_w[j];
        out[0] = s;
    }
}

// ---------------------------------------------------------------------------
extern "C" void kernel_launch(void* const* d_in, const int* in_sizes, int n_in,
                              void* d_out, int out_size, void* d_ws, size_t ws_size,
                              hipStream_t stream) {
    const float* x_seq  = (const float*)d_in[0];   // [T,N,F]
    const int*   ei     = (const int*)d_in[1];     // [T,2,E]
    const float* conv_w = (const float*)d_in[2];   // [F,H]
    const float* conv_b = (const float*)d_in[3];   // [H]
    const float* w_ih   = (const float*)d_in[4];   // [4L,H]
    const float* w_hh   = (const float*)d_in[5];   // [4L,L]
    const float* b_ih   = (const float*)d_in[6];   // [4L]
    const float* b_hh   = (const float*)d_in[7];   // [4L]
    const float* fc_w   = (const float*)d_in[8];   // [1,L]
    const float* fc_b   = (const float*)d_in[9];   // [1]
    float* out = (float*)d_out;

    // workspace carve-up (fp32): dinv [T*N] | xw [T*N*H] | agg [T*N*H] | seq [T*H]
    char* ws = (char*)d_ws;
    float* deg = (float*)ws;                                  // 1.6 MB
    float* xw  = (float*)(ws + ((size_t)T_ * N_ * 4 + 255 & ~(size_t)255));
    float* agg = xw + (size_t)T_ * N_ * H_;                   // +51.2 MB
    float* seq = agg + (size_t)T_ * N_ * H_;                  // +51.2 MB

    const int M = T_ * N_;  // 400000 rows, divisible by 16

    // 0. init
    {
        long tot = (long)T_ * N_ * H_;
        init_kernel<<<(int)((tot + 255) / 256), 256, 0, stream>>>(deg, agg, seq);
    }
    // 1. degree
    {
        long tot = (long)T_ * E_;
        degree_kernel<<<(int)((tot + 255) / 256), 256, 0, stream>>>(ei, deg);
    }
    // 2. rsqrt
    rsqrt_kernel<<<(M + 255) / 256, 256, 0, stream>>>(deg);
    // 3. fp32 WMMA GEMM: xw = x @ conv_w  (8 waves/block, 1 wave per 16-row strip)
    {
        int waves  = M / 16;                  // 25000
        int blocks = (waves + 7) / 8;         // 256 thr = 8 waves
        gemm_xw_wmma<<<blocks, 256, 0, stream>>>(x_seq, conv_w, xw, M);
    }
    // 4. edge scatter (wave per edge, lane per channel)
    {
        long tot = (long)T_ * E_ * 32;
        scatter_kernel<<<(int)((tot + 255) / 256), 256, 0, stream>>>(ei, deg, xw, agg);
    }
    // 5. relu + mean pool
    {
        dim3 grid(128, T_);
        pool_kernel<<<grid, 256, 0, stream>>>(agg, xw, deg, conv_b, seq);
    }
    // 6. LSTM + FC head
    lstm_kernel<<<1, 256, 0, stream>>>(seq, w_ih, w_hh, b_ih, b_hh, fc_w, fc_b, out);
}